// Attention_68264210202738
// MI455X (gfx1250) — compile-verified
//
#include <hip/hip_runtime.h>
#include <hip/hip_bf16.h>
#include <math.h>
#include <stdint.h>

// ---------------------------------------------------------------------------
// Attention block for MI455X (gfx1250), fp32, V_WMMA_F32_16X16X4_F32 path,
// with double-buffered GLOBAL_LOAD_ASYNC_TO_LDS staging (ASYNCcnt-tracked).
// dims: B=2, N=4096, DIM=768, HEADS=12, HEAD_DIM=64, AXIS_DIM=32 (16 freqs)
// ---------------------------------------------------------------------------

typedef float v2f __attribute__((ext_vector_type(2)));
typedef float v8f __attribute__((ext_vector_type(8)));

#define DIMC   768
#define NHEADS 12
#define HDIM   64
#define SEQ    4096
#define BATCH  2

#if defined(__HIP_DEVICE_COMPILE__) &&                                  \
    __has_builtin(__builtin_amdgcn_global_load_async_to_lds_b128) &&    \
    __has_builtin(__builtin_amdgcn_s_wait_asynccnt)
#define USE_ASYNC_LDS 1
#else
#define USE_ASYNC_LDS 0
#endif

#if USE_ASYNC_LDS
// Matches the builtin's parameter type: gcc vector of 4 ints,
// AS1 (global) for the source, AS3 (shared) for the destination.
typedef int v4i_raw __attribute__((vector_size(16)));
typedef __attribute__((address_space(1))) v4i_raw* as1_v4i;
typedef __attribute__((address_space(3))) v4i_raw* as3_v4i;
#endif

__device__ __forceinline__ void copy16_to_lds(float* lds_dst,
                                              const float* gsrc) {
#if USE_ASYNC_LDS
  // Per-lane async DMA: 16B global -> LDS, tracked by ASYNCcnt.
  // generic->AS1 is numerically identity; a generic LDS pointer's low 32 bits
  // are the LDS byte offset (ISA 10.2), so route casts through integers.
  __builtin_amdgcn_global_load_async_to_lds_b128(
      (as1_v4i)(uintptr_t)gsrc,
      (as3_v4i)(unsigned)(uintptr_t)lds_dst,
      /*offset=*/0, /*cpol=*/0);
#else
  *(float4*)lds_dst = *(const float4*)gsrc;
#endif
}

__device__ __forceinline__ void wait_async_copies() {
#if USE_ASYNC_LDS
  __builtin_amdgcn_s_wait_asynccnt(0);
#endif
}

__device__ __forceinline__ v8f wmma4(v2f a, v2f b, v8f c) {
  // D = A(16x4) * B(4x16) + C(16x16), fp32 exact
  return __builtin_amdgcn_wmma_f32_16x16x4_f32(
      /*neg_a=*/false, a, /*neg_b=*/false, b,
      /*c_mod=*/(short)0, c, /*reuse_a=*/false, /*reuse_b=*/false);
}

// ---------------------------------------------------------------------------
// Generic GEMM: out(M x N) = A(M x K) * W(N x K)^T + bias
// block = 128 threads (4 waves), 64x64 output tile, double-buffered LDS.
// mode 0: out[row*768 + col]
// mode 1: scatter q -> qd[b][h][n][d]
// mode 2: scatter kv -> kd / vd [b][h][n][d]
// ---------------------------------------------------------------------------
__global__ __launch_bounds__(128) void gemm_wmma_f32(
    const float* __restrict__ A, const float* __restrict__ W,
    const float* __restrict__ bias, float* __restrict__ out,
    int K, int mode, float* __restrict__ qd, float* __restrict__ kd,
    float* __restrict__ vd) {
  __shared__ float A_lds[2][64][36];  // stride 36: 16 distinct banks over rows
  __shared__ float W_lds[2][64][36];

  const int tid = threadIdx.x;
  const int wv = tid >> 5;        // wave 0..3
  const int l = tid & 31;
  const int lh = l >> 4;          // lane half (K-pair select)
  const int ll = l & 15;          // lane in half (M or N index)

  const int n0 = blockIdx.x * 64;
  const int m0 = blockIdx.y * 64;

  v8f acc[4] = {};

  const int row_ld = tid >> 1;            // 0..63
  const int col_ld = (tid & 1) * 16;      // 0 or 16

  // stage one 64x32 tile of A and of W into buffer `buf`
  auto stage = [&](int buf, int k0) {
    const float* srcA = A + (size_t)(m0 + row_ld) * K + k0 + col_ld;
    const float* srcW = W + (size_t)(n0 + row_ld) * K + k0 + col_ld;
    float* dstA = &A_lds[buf][row_ld][col_ld];
    float* dstW = &W_lds[buf][row_ld][col_ld];
#pragma unroll
    for (int q = 0; q < 16; q += 4) {
      copy16_to_lds(dstA + q, srcA + q);
      copy16_to_lds(dstW + q, srcW + q);
    }
  };

  stage(0, 0);
  const int nchunks = K / 32;
  for (int it = 0; it < nchunks; ++it) {
    const int buf = it & 1;
    wait_async_copies();
    __syncthreads();
    if (it + 1 < nchunks) stage(buf ^ 1, (it + 1) * 32);

#pragma unroll
    for (int kk = 0; kk < 32; kk += 4) {
      v2f a = *(const v2f*)&A_lds[buf][wv * 16 + ll][kk + 2 * lh];
#pragma unroll
      for (int nt = 0; nt < 4; ++nt) {
        v2f b = *(const v2f*)&W_lds[buf][nt * 16 + ll][kk + 2 * lh];
        acc[nt] = wmma4(a, b, acc[nt]);
      }
    }
  }

#pragma unroll
  for (int nt = 0; nt < 4; ++nt) {
#pragma unroll
    for (int i = 0; i < 8; ++i) {
      const int row = m0 + wv * 16 + i + 8 * lh;
      const int colG = n0 + nt * 16 + ll;
      float val = acc[nt][i] + (bias ? bias[colG] : 0.0f);
      if (mode == 0) {
        out[(size_t)row * DIMC + colG] = val;
      } else if (mode == 1) {
        const int h = colG >> 6, d = colG & 63;
        const int b = row >> 12, n = row & (SEQ - 1);
        qd[((((size_t)b * NHEADS) + h) * SEQ + n) * HDIM + d] = val;
      } else {
        const int s = colG >= DIMC;
        const int rem = colG - (s ? DIMC : 0);
        const int h = rem >> 6, d = rem & 63;
        const int b = row >> 12, n = row & (SEQ - 1);
        float* dst = s ? vd : kd;
        dst[((((size_t)b * NHEADS) + h) * SEQ + n) * HDIM + d] = val;
      }
    }
  }
}

// ---------------------------------------------------------------------------
// Per-head RMS norm + 2D rotary, in place on qd and kd.
// One wave per (b,h,n) row: lane l owns dims {2l, 2l+1}; axis = l>=16,
// freq index j = l&15 -> matches rope even/odd pairing exactly.
// ---------------------------------------------------------------------------
__global__ __launch_bounds__(128) void normrope_kernel(
    float* __restrict__ qd, float* __restrict__ kd,
    const int* __restrict__ positions, const float* __restrict__ qw,
    const float* __restrict__ kw, const float* __restrict__ invf) {
  const int wv = threadIdx.x >> 5;
  const int l = threadIdx.x & 31;
  const int r = blockIdx.x * 4 + wv;  // 0 .. B*H*SEQ-1
  const int b = r / (NHEADS * SEQ);
  const int rem = r - b * (NHEADS * SEQ);
  const int h = rem / SEQ;
  const int n = rem - h * SEQ;

  const int axis = l >> 4;
  const int j = l & 15;
  const int pos = positions[((size_t)(b * SEQ + n)) * 2 + axis];
  const bool valid = pos >= 0;
  const float p = (float)(pos < 0 ? 0 : pos);
  const float ang = p * invf[j];
  float sn, cs;
  __sincosf(ang, &sn, &cs);

  const size_t base = ((((size_t)b * NHEADS) + h) * SEQ + n) * HDIM;

#pragma unroll
  for (int which = 0; which < 2; ++which) {
    float* ptr = which ? kd : qd;
    const float* w = which ? kw : qw;
    v2f x2 = *(v2f*)&ptr[base + 2 * l];
    // RMS over 64 dims: full-wave reduction (wave32)
    float ss = x2.x * x2.x + x2.y * x2.y;
#pragma unroll
    for (int off = 1; off < 32; off <<= 1) ss += __shfl_xor(ss, off, 32);
    const float inv = rsqrtf(ss * (1.0f / 64.0f) + 1e-6f);
    float e = x2.x * inv * w[2 * l];
    float o = x2.y * inv * w[2 * l + 1];
    float re = e * cs - o * sn;
    float ro = e * sn + o * cs;
    if (!valid) { re = e; ro = o; }
    v2f out2; out2.x = re; out2.y = ro;
    *(v2f*)&ptr[base + 2 * l] = out2;
  }
}

// ---------------------------------------------------------------------------
// Flash attention: block = 128 threads (4 waves) owns a 64-row Q tile of one
// (b,h). Streams K/V in double-buffered 64x64 LDS tiles (async DMA), online
// softmax, O += P*V via WMMA. Writes o in (b, n, h*64+d) layout so the final
// projection is a plain GEMM.
// ---------------------------------------------------------------------------
__global__ __launch_bounds__(128) void attn_kernel(
    const float* __restrict__ qd, const float* __restrict__ kd,
    const float* __restrict__ vd, float* __restrict__ omat) {
  __shared__ float K_lds[2][64][68];
  __shared__ float V_lds[2][64][68];
  __shared__ float P_lds[4][16][68];

  const int tid = threadIdx.x;
  const int wv = tid >> 5;
  const int l = tid & 31;
  const int lh = l >> 4;
  const int ll = l & 15;

  const int qb = blockIdx.x;   // q tile (64 rows)
  const int h = blockIdx.y;
  const int b = blockIdx.z;

  const size_t headbase = (((size_t)b * NHEADS) + h) * SEQ * HDIM;

  // Preload this wave's Q fragments (A-layout), scale folded in (1/sqrt(64)).
  const int qrow = qb * 64 + wv * 16 + ll;
  v2f qp[16];
#pragma unroll
  for (int t = 0; t < 16; ++t) {
    v2f v = *(const v2f*)&qd[headbase + (size_t)qrow * HDIM + 4 * t + 2 * lh];
    qp[t] = v * 0.125f;
  }

  v8f o[4] = {};
  float mrow[8], lrow[8];
#pragma unroll
  for (int i = 0; i < 8; ++i) { mrow[i] = -INFINITY; lrow[i] = 0.0f; }

  const int row_ld = tid >> 1;
  const int col_ld = (tid & 1) * 32;

  auto stageKV = [&](int buf, int kt) {
    const float* ksrc =
        kd + headbase + (size_t)(kt * 64 + row_ld) * HDIM + col_ld;
    const float* vsrc =
        vd + headbase + (size_t)(kt * 64 + row_ld) * HDIM + col_ld;
    float* kdst = &K_lds[buf][row_ld][col_ld];
    float* vdst = &V_lds[buf][row_ld][col_ld];
#pragma unroll
    for (int q = 0; q < 32; q += 4) {
      copy16_to_lds(kdst + q, ksrc + q);
      copy16_to_lds(vdst + q, vsrc + q);
    }
  };

  stageKV(0, 0);
  const int nchunks = SEQ / 64;
  for (int kt = 0; kt < nchunks; ++kt) {
    const int buf = kt & 1;
    wait_async_copies();
    __syncthreads();
    if (kt + 1 < nchunks) stageKV(buf ^ 1, kt + 1);

    // S = Q * K^T  (16x64 per wave)
    v8f s[4] = {};
#pragma unroll
    for (int t = 0; t < 16; ++t) {
      const int kk = 4 * t;
      v2f a = qp[t];
#pragma unroll
      for (int nt = 0; nt < 4; ++nt) {
        v2f bb = *(const v2f*)&K_lds[buf][nt * 16 + ll][kk + 2 * lh];
        s[nt] = wmma4(a, bb, s[nt]);
      }
    }

    // online softmax; row r = i + 8*lh lives in lane-half {lh}
#pragma unroll
    for (int i = 0; i < 8; ++i) {
      float tm = fmaxf(fmaxf(s[0][i], s[1][i]), fmaxf(s[2][i], s[3][i]));
#pragma unroll
      for (int off = 1; off < 16; off <<= 1)
        tm = fmaxf(tm, __shfl_xor(tm, off, 32));
      const float mn = fmaxf(mrow[i], tm);
      const float corr = __expf(mrow[i] - mn);
      float rs = 0.0f;
#pragma unroll
      for (int nt = 0; nt < 4; ++nt) {
        s[nt][i] = __expf(s[nt][i] - mn);
        rs += s[nt][i];
      }
#pragma unroll
      for (int off = 1; off < 16; off <<= 1) rs += __shfl_xor(rs, off, 32);
      lrow[i] = lrow[i] * corr + rs;
      mrow[i] = mn;
#pragma unroll
      for (int dt = 0; dt < 4; ++dt) o[dt][i] *= corr;
    }

    // P: C-layout -> LDS -> A-layout (per-wave region, wave-local ordering)
#pragma unroll
    for (int nt = 0; nt < 4; ++nt)
#pragma unroll
      for (int i = 0; i < 8; ++i)
        P_lds[wv][i + 8 * lh][nt * 16 + ll] = s[nt][i];

    // O += P * V
#pragma unroll
    for (int t = 0; t < 16; ++t) {
      const int kk = 4 * t;
      v2f a = *(const v2f*)&P_lds[wv][ll][kk + 2 * lh];
#pragma unroll
      for (int dt = 0; dt < 4; ++dt) {
        v2f bb;
        bb.x = V_lds[buf][kk + 2 * lh][dt * 16 + ll];
        bb.y = V_lds[buf][kk + 2 * lh + 1][dt * 16 + ll];
        o[dt] = wmma4(a, bb, o[dt]);
      }
    }
  }

  // normalize and store o in (b, n, 768) layout
#pragma unroll
  for (int i = 0; i < 8; ++i) {
    const float inv = 1.0f / lrow[i];
    const int row = qb * 64 + wv * 16 + i + 8 * lh;
#pragma unroll
    for (int dt = 0; dt < 4; ++dt) {
      omat[((size_t)(b * SEQ + row)) * DIMC + h * HDIM + dt * 16 + ll] =
          o[dt][i] * inv;
    }
  }
}

// ---------------------------------------------------------------------------
// Inputs (setup_inputs order):
// 0:x 1:positions 2:q_w 3:q_b 4:kv_w 5:kv_b 6:proj_w 7:proj_b
// 8:q_norm_w 9:k_norm_w 10:inv_freq
// Workspace: qd | kd | vd | omat, each 8192*768 floats (~101 MB total).
// ---------------------------------------------------------------------------
extern "C" void kernel_launch(void* const* d_in, const int* in_sizes, int n_in,
                              void* d_out, int out_size, void* d_ws,
                              size_t ws_size, hipStream_t stream) {
  const float* x = (const float*)d_in[0];
  const int* positions = (const int*)d_in[1];
  const float* q_w = (const float*)d_in[2];
  const float* q_b = (const float*)d_in[3];
  const float* kv_w = (const float*)d_in[4];
  const float* kv_b = (const float*)d_in[5];
  const float* proj_w = (const float*)d_in[6];
  const float* proj_b = (const float*)d_in[7];
  const float* q_norm_w = (const float*)d_in[8];
  const float* k_norm_w = (const float*)d_in[9];
  const float* inv_freq = (const float*)d_in[10];

  float* ws = (float*)d_ws;
  const size_t per = (size_t)BATCH * SEQ * DIMC;  // 6,291,456 floats
  float* qd = ws;
  float* kd = ws + per;
  float* vd = ws + 2 * per;
  float* omat = ws + 3 * per;
  float* out = (float*)d_out;

  const int M = BATCH * SEQ;  // 8192

  // 1) Q projection: x @ q_w^T + q_b, scatter to qd[b][h][n][d]
  gemm_wmma_f32<<<dim3(DIMC / 64, M / 64), 128, 0, stream>>>(
      x, q_w, q_b, nullptr, DIMC, 1, qd, nullptr, nullptr);
  // 2) KV projection: x @ kv_w.T + kv_b, scatter to kd/vd
  gemm_wmma_f32<<<dim3(2 * DIMC / 64, M / 64), 128, 0, stream>>>(
      x, kv_w, kv_b, nullptr, DIMC, 2, nullptr, kd, vd);
  // 3) RMS norm + 2D rope in place on qd, kd
  normrope_kernel<<<(BATCH * NHEADS * SEQ) / 4, 128, 0, stream>>>(
      qd, kd, positions, q_norm_w, k_norm_w, inv_freq);
  // 4) Flash attention -> omat (b, n, 768)
  attn_kernel<<<dim3(SEQ / 64, NHEADS, BATCH), 128, 0, stream>>>(qd, kd, vd,
                                                                 omat);
  // 5) Output projection: omat @ proj_w^T + proj_b -> d_out
  gemm_wmma_f32<<<dim3(DIMC / 64, M / 64), 128, 0, stream>>>(
      omat, proj_w, proj_b, out, DIMC, 0, nullptr, nullptr, nullptr);
}